// CosineAttention_52656299049099
// MI455X (gfx1250) — compile-verified
//
#include <hip/hip_runtime.h>
#include <math.h>

typedef __attribute__((ext_vector_type(2))) float v2f;
typedef __attribute__((ext_vector_type(8))) float v8f;

#define BATCH 64
#define LKEYS 4096
#define DDIM  1024
#define EPSN  1e-12f

// ---------------------------------------------------------------------------
// Kernel 0: q_hat[b,:] = q[b,:] / max(||q[b,:]||, eps)   (64 x 1024, tiny)
// ---------------------------------------------------------------------------
__global__ void cosattn_qnorm(const float* __restrict__ q, float* __restrict__ qn) {
    __shared__ float red[256];
    const int b = blockIdx.x;
    const int t = threadIdx.x;
    const float* row = q + (size_t)b * DDIM;
    float ss = 0.0f;
    for (int i = t; i < DDIM; i += 256) {
        float v = row[i];
        ss += v * v;
    }
    red[t] = ss;
    __syncthreads();
    for (int s = 128; s > 0; s >>= 1) {
        if (t < s) red[t] += red[t + s];
        __syncthreads();
    }
    const float rn = 1.0f / fmaxf(sqrtf(red[0]), EPSN);
    float* outr = qn + (size_t)b * DDIM;
    for (int i = t; i < DDIM; i += 256) outr[i] = row[i] * rn;
}

// ---------------------------------------------------------------------------
// Kernel 1: raw scores via V_WMMA_F32_16X16X4_F32.
// One wave32 per 16-row key tile. A = 16x4 key chunk (ISA A-layout),
// B = q_hat chunk replicated across all 16 columns (C/D-style half-lane K
// split), so every accumulator column holds the same dot product.
// Self-dot k.k accumulated in VALU alongside, reduced with shfl_xor(16).
// ---------------------------------------------------------------------------
__global__ void cosattn_scores(const float* __restrict__ keys,
                               const float* __restrict__ qn,
                               float* __restrict__ out) {
    const int gtid = blockIdx.x * blockDim.x + threadIdx.x;
    const int wave = gtid >> 5;                 // 0 .. 64*256-1
    const int lane = threadIdx.x & 31;

    const int b    = wave >> 8;                 // batch (256 tiles per batch)
    const int tile = wave & 255;                // tile within batch
    const int row  = (tile << 4) + (lane & 15); // this lane's key row (16 rows/tile)
    const int koff = (lane >> 4) << 1;          // lanes 0-15: k%4 in {0,1}; 16-31: {2,3}

    const float* krow = keys + ((size_t)b * LKEYS + row) * DDIM;
    const float* qrow = qn + (size_t)b * DDIM;

    v8f acc0 = {};
    v8f acc1 = {};
    float ss = 0.0f;

#pragma unroll 4
    for (int c = 0; c < 256; c += 2) {
        const int k0 = c * 4 + koff;
        v2f a0 = *(const v2f*)(krow + k0);
        v2f b0 = *(const v2f*)(qrow + k0);
        v2f a1 = *(const v2f*)(krow + k0 + 4);
        v2f b1 = *(const v2f*)(qrow + k0 + 4);
        acc0 = __builtin_amdgcn_wmma_f32_16x16x4_f32(
            false, a0, false, b0, (short)0, acc0, false, false);
        acc1 = __builtin_amdgcn_wmma_f32_16x16x4_f32(
            false, a1, false, b1, (short)0, acc1, false, false);
        ss += a0.x * a0.x + a0.y * a0.y + a1.x * a1.x + a1.y * a1.y;
    }
    v8f acc = acc0 + acc1;

    // full sum-of-squares for this lane's row (k%4 {0,1} half + {2,3} half)
    ss += __shfl_xor(ss, 16, 32);

    // Extract dot[row]: VGPR v holds M=v (lanes 0-15) / M=v+8 (lanes 16-31);
    // all columns are identical.
    const int k7 = lane & 7;
    float h = acc[0];
    h = (k7 == 1) ? acc[1] : h;
    h = (k7 == 2) ? acc[2] : h;
    h = (k7 == 3) ? acc[3] : h;
    h = (k7 == 4) ? acc[4] : h;
    h = (k7 == 5) ? acc[5] : h;
    h = (k7 == 6) ? acc[6] : h;
    h = (k7 == 7) ? acc[7] : h;
    const int r = lane & 15;
    const int src = r + ((r & 8) << 1);   // r<8 -> lane r ; r>=8 -> lane r+16
    const float dot = __shfl(h, src, 32);

    if (lane < 16) {
        const float score = dot / fmaxf(sqrtf(ss), EPSN);
        out[(size_t)b * LKEYS + row] = score;
    }
}

// ---------------------------------------------------------------------------
// Kernel 2: masked softmax over L=4096 per batch, in place on d_out.
// ---------------------------------------------------------------------------
__global__ void cosattn_softmax(const unsigned char* __restrict__ mask,
                                float* __restrict__ out) {
    __shared__ float red[256];
    const int b = blockIdx.x;
    const int t = threadIdx.x;
    float* row = out + (size_t)b * LKEYS;
    const unsigned char* mrow = mask + (size_t)b * LKEYS;

    float vals[16];
    float mx = -INFINITY;
#pragma unroll
    for (int i = 0; i < 16; ++i) {
        const int idx = t + i * 256;
        const float v = mrow[idx] ? row[idx] : -INFINITY;
        vals[i] = v;
        mx = fmaxf(mx, v);
    }
    red[t] = mx;
    __syncthreads();
    for (int s = 128; s > 0; s >>= 1) {
        if (t < s) red[t] = fmaxf(red[t], red[t + s]);
        __syncthreads();
    }
    mx = red[0];
    __syncthreads();

    float sum = 0.0f;
#pragma unroll
    for (int i = 0; i < 16; ++i) {
        const float e = expf(vals[i] - mx);
        vals[i] = e;
        sum += e;
    }
    red[t] = sum;
    __syncthreads();
    for (int s = 128; s > 0; s >>= 1) {
        if (t < s) red[t] += red[t + s];
        __syncthreads();
    }
    const float inv = 1.0f / red[0];
#pragma unroll
    for (int i = 0; i < 16; ++i) row[t + i * 256] = vals[i] * inv;
}

// ---------------------------------------------------------------------------
extern "C" void kernel_launch(void* const* d_in, const int* in_sizes, int n_in,
                              void* d_out, int out_size, void* d_ws, size_t ws_size,
                              hipStream_t stream) {
    const float* q            = (const float*)d_in[0];   // [64,1024] f32
    const float* keys         = (const float*)d_in[1];   // [64,4096,1024] f32
    const unsigned char* mask = (const unsigned char*)d_in[2]; // [64,4096] bool
    float* out = (float*)d_out;                          // [64,4096] f32
    float* qn  = (float*)d_ws;                           // 64*1024 f32 = 256 KB

    cosattn_qnorm<<<BATCH, 256, 0, stream>>>(q, qn);

    // 64 batches * 256 tiles = 16384 waves; 8 waves (256 threads) per block.
    cosattn_scores<<<(BATCH * (LKEYS / 16)) / 8, 256, 0, stream>>>(keys, qn, out);

    cosattn_softmax<<<BATCH, 256, 0, stream>>>(mask, out);
}